// MultiHeadAttentionQuantum_65481071409700
// MI455X (gfx1250) — compile-verified
//
#include <hip/hip_runtime.h>

// Fused MHA-Quantum kernel for MI455X (gfx1250, wave32).
//
// Reference collapses analytically: k,v are dead; the 8-wire quantum layer on
// q = x@Wq.T + bq reduces (product state + CNOT ring = bit-XOR relabeling) to
//   c_j  = cos(q_j + phi_j)
//   o[0] = c1*...*c7 ;  o[w] = c0*...*cw (w>=1)
// followed by o@Wc.T + bc.  Total HBM traffic 2 MB -> ~90 ns roofline floor;
// purely bandwidth bound, so everything is fused in one pass.
// Both 8x8 projections run on V_WMMA_F32_16X16X4_F32 (K=8 as two K=4 steps,
// N padded to 16 with zeroed B columns).

typedef float v2f __attribute__((ext_vector_type(2)));
typedef float v8f __attribute__((ext_vector_type(8)));

#define TOK_PER_BLOCK 256
#define TOK_PER_WAVE  32
#define NWAVES        8
#define EDIM          8

__global__ __launch_bounds__(256)
void mhaq_fused_kernel(const float* __restrict__ x,
                       const float* __restrict__ Wq,
                       const float* __restrict__ bq,
                       const float* __restrict__ Wc,
                       const float* __restrict__ bc,
                       const float* __restrict__ phi,
                       float* __restrict__ out,
                       int nTokens)
{
    // padded to 9 floats/row: lane-stride 9 dwords is conflict-free mod 64 banks
    __shared__ float sq[NWAVES][TOK_PER_WAVE][9];
    __shared__ float so[NWAVES][TOK_PER_WAVE][9];

    const int lane = threadIdx.x & 31;
    const int wave = threadIdx.x >> 5;
    const int col  = lane & 15;          // A-row / B-column / D-column index
    const int kb   = (lane >> 4) << 1;   // K pair base: lanes 0-15 -> K{0,1}, 16-31 -> K{2,3}
    const bool act = (col < EDIM);       // only 8 of 16 padded columns are real

    // Block-uniform guard (launch config guarantees nTokens % 256 == 0)
    if (blockIdx.x * TOK_PER_BLOCK >= nTokens) return;
    const int tokBase = blockIdx.x * TOK_PER_BLOCK + wave * TOK_PER_WAVE;

    // ---- B operands: y = x @ W.T  =>  B[k][n] = W[n][k] (contiguous in W row n)
    // 4x16 f32 B layout: VGPR0 = {K=kb}, VGPR1 = {K=kb+1}, per-lane column n=col.
    v2f BqLo = {0.f, 0.f}, BqHi = {0.f, 0.f};
    v2f BcLo = {0.f, 0.f}, BcHi = {0.f, 0.f};
    float addQ = 0.f, addC = 0.f;
    if (act) {
        BqLo = *(const v2f*)(Wq + col * EDIM + kb);
        BqHi = *(const v2f*)(Wq + col * EDIM + kb + 4);
        BcLo = *(const v2f*)(Wc + col * EDIM + kb);
        BcHi = *(const v2f*)(Wc + col * EDIM + kb + 4);
        addQ = bq[col];
        addC = bc[col];
    }

    float ph[EDIM];
    #pragma unroll
    for (int j = 0; j < EDIM; ++j) ph[j] = phi[j];

    // ---- A tiles: 16x4 f32 A layout = float2 per lane, row M = col ----
    const float* r0 = x + (size_t)(tokBase + col) * EDIM;
    const float* r1 = x + (size_t)(tokBase + 16 + col) * EDIM;
    v2f a0lo = *(const v2f*)(r0 + kb);
    v2f a0hi = *(const v2f*)(r0 + kb + 4);
    v2f a1lo = *(const v2f*)(r1 + kb);
    v2f a1hi = *(const v2f*)(r1 + kb + 4);

    // ---- q = x @ Wq.T  (two K=4 steps per 16-token tile) ----
    v8f q0 = {}, q1 = {};
    q0 = __builtin_amdgcn_wmma_f32_16x16x4_f32(false, a0lo, false, BqLo, (short)0, q0, false, false);
    q0 = __builtin_amdgcn_wmma_f32_16x16x4_f32(false, a0hi, false, BqHi, (short)0, q0, false, false);
    q1 = __builtin_amdgcn_wmma_f32_16x16x4_f32(false, a1lo, false, BqLo, (short)0, q1, false, false);
    q1 = __builtin_amdgcn_wmma_f32_16x16x4_f32(false, a1hi, false, BqHi, (short)0, q1, false, false);

    // D layout: VGPR r, lanes 0-15 -> M=r, lanes 16-31 -> M=r+8; column = col.
    if (act) {
        const int mb = (lane >> 4) << 3;
        #pragma unroll
        for (int r = 0; r < 8; ++r) {
            sq[wave][mb + r][col]      = q0[r] + addQ;
            sq[wave][16 + mb + r][col] = q1[r] + addQ;
        }
    }
    __syncthreads();

    // ---- quantum layer: one token per lane (closed form) ----
    {
        float c[EDIM];
        #pragma unroll
        for (int j = 0; j < EDIM; ++j)
            c[j] = __cosf(sq[wave][lane][j] + ph[j]);   // v_cos_f32 path

        float o[EDIM];
        float p = c[0];
        #pragma unroll
        for (int w = 1; w < EDIM; ++w) { p *= c[w]; o[w] = p; }  // o[w] = c0..cw
        float u = c[1];
        #pragma unroll
        for (int j = 2; j < EDIM; ++j) u *= c[j];
        o[0] = u;                                                 // o[0] = c1..c7

        #pragma unroll
        for (int j = 0; j < EDIM; ++j) so[wave][lane][j] = o[j];
    }
    __syncthreads();

    // ---- final = o @ Wc.T + bc ----
    v2f A0lo = { so[wave][col][kb],        so[wave][col][kb + 1] };
    v2f A0hi = { so[wave][col][kb + 4],    so[wave][col][kb + 5] };
    v2f A1lo = { so[wave][16 + col][kb],   so[wave][16 + col][kb + 1] };
    v2f A1hi = { so[wave][16 + col][kb + 4], so[wave][16 + col][kb + 5] };

    v8f d0 = {}, d1 = {};
    d0 = __builtin_amdgcn_wmma_f32_16x16x4_f32(false, A0lo, false, BcLo, (short)0, d0, false, false);
    d0 = __builtin_amdgcn_wmma_f32_16x16x4_f32(false, A0hi, false, BcHi, (short)0, d0, false, false);
    d1 = __builtin_amdgcn_wmma_f32_16x16x4_f32(false, A1lo, false, BcLo, (short)0, d1, false, false);
    d1 = __builtin_amdgcn_wmma_f32_16x16x4_f32(false, A1hi, false, BcHi, (short)0, d1, false, false);

    if (act) {
        const int mb = (lane >> 4) << 3;
        #pragma unroll
        for (int r = 0; r < 8; ++r) {
            sq[wave][mb + r][col]      = d0[r] + addC;
            sq[wave][16 + mb + r][col] = d1[r] + addC;
        }
    }
    __syncthreads();

    // ---- coalesced store: one token per lane, two b128 stores ----
    float vrow[EDIM];
    #pragma unroll
    for (int j = 0; j < EDIM; ++j) vrow[j] = sq[wave][lane][j];
    float* dst = out + (size_t)(tokBase + lane) * EDIM;
    *(float4*)(dst)     = make_float4(vrow[0], vrow[1], vrow[2], vrow[3]);
    *(float4*)(dst + 4) = make_float4(vrow[4], vrow[5], vrow[6], vrow[7]);
}

extern "C" void kernel_launch(void* const* d_in, const int* in_sizes, int n_in,
                              void* d_out, int out_size, void* d_ws, size_t ws_size,
                              hipStream_t stream) {
    // setup_inputs order: x, Wq, bq, Wk, bk, Wv, bv, Wc, bc, phi (k/v dead)
    const float* x   = (const float*)d_in[0];
    const float* Wq  = (const float*)d_in[1];
    const float* bq  = (const float*)d_in[2];
    const float* Wc  = (const float*)d_in[7];
    const float* bc  = (const float*)d_in[8];
    const float* phi = (const float*)d_in[9];
    float* out = (float*)d_out;

    const int nTokens = in_sizes[0] / EDIM;                    // B*S = 32768
    const int grid    = (nTokens + TOK_PER_BLOCK - 1) / TOK_PER_BLOCK;  // 128
    mhaq_fused_kernel<<<grid, 256, 0, stream>>>(x, Wq, bq, Wc, bc, phi, out, nTokens);
}